// m2_model_90134183674372
// MI455X (gfx1250) — compile-verified
//
#include <hip/hip_runtime.h>
#include <hip/hip_bf16.h>
#include <math.h>

typedef __attribute__((ext_vector_type(2))) float v2f;
typedef __attribute__((ext_vector_type(8))) float v8f;

#define WAVE 32

// ---------------------------------------------------------------------------
// Utility kernels
// ---------------------------------------------------------------------------
__global__ void fill_val_kernel(float* __restrict__ p, size_t n, float v) {
    size_t stride = (size_t)gridDim.x * blockDim.x;
    for (size_t i = (size_t)blockIdx.x * blockDim.x + threadIdx.x; i < n; i += stride)
        p[i] = v;
}

__global__ void add_kernel(float* __restrict__ a, const float* __restrict__ b, size_t n) {
    size_t stride = (size_t)gridDim.x * blockDim.x;
    for (size_t i = (size_t)blockIdx.x * blockDim.x + threadIdx.x; i < n; i += stride)
        a[i] += b[i];
}

// p[i,f] = max(p[i,f] + bias[f], 0)
__global__ void bias_relu_kernel(float* __restrict__ p, const float* __restrict__ bias,
                                 size_t n, int F) {
    size_t stride = (size_t)gridDim.x * blockDim.x;
    for (size_t i = (size_t)blockIdx.x * blockDim.x + threadIdx.x; i < n; i += stride)
        p[i] = fmaxf(p[i] + bias[i % F], 0.0f);
}

// out[i,f] = bias[f]
__global__ void init_bias_kernel(float* __restrict__ p, const float* __restrict__ bias,
                                 size_t n, int F) {
    size_t stride = (size_t)gridDim.x * blockDim.x;
    for (size_t i = (size_t)blockIdx.x * blockDim.x + threadIdx.x; i < n; i += stride)
        p[i] = bias[i % F];
}

// agg[i,f] /= max(cnt[i],1)
__global__ void mean_div_kernel(float* __restrict__ agg, const float* __restrict__ cnt,
                                size_t n, int F) {
    size_t stride = (size_t)gridDim.x * blockDim.x;
    for (size_t i = (size_t)blockIdx.x * blockDim.x + threadIdx.x; i < n; i += stride)
        agg[i] /= fmaxf(cnt[i / F], 1.0f);
}

__global__ void dinv_kernel(float* __restrict__ dinv, const float* __restrict__ cnt, int n) {
    int i = blockIdx.x * blockDim.x + threadIdx.x;
    if (i < n) dinv[i] = rsqrtf(cnt[i] + 1.0f);  // degree includes the self-loop
}

// ---------------------------------------------------------------------------
// Degree count (original edges only)
// ---------------------------------------------------------------------------
__global__ void degree_kernel(const long long* __restrict__ dst, int E, float* __restrict__ cnt) {
    size_t stride = (size_t)gridDim.x * blockDim.x;
    for (size_t e = (size_t)blockIdx.x * blockDim.x + threadIdx.x; e < (size_t)E; e += stride)
        atomicAdd(&cnt[(int)dst[e]], 1.0f);
}

// out[dst[e], f] += feat[src[e], f]
__global__ void edge_scatter_sum_kernel(const long long* __restrict__ src,
                                        const long long* __restrict__ dst, int E,
                                        const float* __restrict__ feat,
                                        float* __restrict__ out, int F) {
    size_t total = (size_t)E * F;
    size_t stride = (size_t)gridDim.x * blockDim.x;
    for (size_t i = (size_t)blockIdx.x * blockDim.x + threadIdx.x; i < total; i += stride) {
        int e = (int)(i / F);
        int f = (int)(i % F);
        int s = (int)src[e], d = (int)dst[e];
        atomicAdd(&out[(size_t)d * F + f], feat[(size_t)s * F + f]);
    }
}

// ---------------------------------------------------------------------------
// FP32 WMMA GEMM: C[nrows, kout] (+)= A[nrows, kin] @ W[kin, kout] (+ bias) (relu)
// One wave computes a 32x32 output tile (2x2 grid of 16x16 WMMA tiles):
// per K-step (4): 2 A fragments + 2 B fragments feed 4 v_wmma_f32_16x16x4_f32,
// halving bytes/FLOP vs one-tile-per-wave.
// A frag: lane L holds A[m, 2*(L/16)+j], m = L%16, j = 0..1     (16x4, 2 VGPRs)
// B frag: lane L holds W[2*(L/16)+j, n], n = L%16               (4x16, 2 VGPRs)
// D frag: 8 VGPRs; m = r + 8*(L/16), n = L%16
// ---------------------------------------------------------------------------
__global__ void gemm_f32_wmma_kernel(const float* __restrict__ A,
                                     const float* __restrict__ W,
                                     const float* __restrict__ bias,
                                     float* __restrict__ C,
                                     int nrows, int kin, int kout,
                                     int accum, int relu) {
    int wave = blockIdx.x * (blockDim.x / WAVE) + (threadIdx.x / WAVE);
    int lane = threadIdx.x & (WAVE - 1);
    int tiles_n = kout >> 5;   // 32-wide tiles
    int tiles_m = nrows >> 5;  // 32-tall tiles
    if (wave >= tiles_m * tiles_n) return;           // wave-uniform: EXEC all-ones for WMMA
    int row0 = (wave / tiles_n) * 32;
    int col0 = (wave % tiles_n) * 32;

    int mn = lane & 15;             // A row within 16-tile / B,C column within 16-tile
    int khalf = (lane >> 4) * 2;    // k sub-offset: 0 or 2

    v8f acc00 = {0.f,0.f,0.f,0.f,0.f,0.f,0.f,0.f};
    v8f acc01 = acc00, acc10 = acc00, acc11 = acc00;

    const float* Arow0 = A + (size_t)(row0 + mn) * kin;        // rows row0..row0+15
    const float* Arow1 = Arow0 + (size_t)16 * kin;             // rows row0+16..+31
    const float* Bc0   = W + col0 + mn;                        // cols col0..+15
    const float* Bc1   = Bc0 + 16;                             // cols col0+16..+31

    for (int k0 = 0; k0 < kin; k0 += 4) {
        int kk = k0 + khalf;
        v2f a0, a1, b0, b1;
        a0.x = Arow0[kk];  a0.y = Arow0[kk + 1];
        a1.x = Arow1[kk];  a1.y = Arow1[kk + 1];
        b0.x = Bc0[(size_t)kk * kout];  b0.y = Bc0[(size_t)(kk + 1) * kout];
        b1.x = Bc1[(size_t)kk * kout];  b1.y = Bc1[(size_t)(kk + 1) * kout];
        acc00 = __builtin_amdgcn_wmma_f32_16x16x4_f32(false, a0, false, b0, (short)0, acc00, false, false);
        acc01 = __builtin_amdgcn_wmma_f32_16x16x4_f32(false, a0, false, b1, (short)0, acc01, false, false);
        acc10 = __builtin_amdgcn_wmma_f32_16x16x4_f32(false, a1, false, b0, (short)0, acc10, false, false);
        acc11 = __builtin_amdgcn_wmma_f32_16x16x4_f32(false, a1, false, b1, (short)0, acc11, false, false);
    }

    int mbase = (lane >> 4) * 8;
    int col_g0 = col0 + mn, col_g1 = col0 + 16 + mn;
    float bv0 = bias ? bias[col_g0] : 0.0f;
    float bv1 = bias ? bias[col_g1] : 0.0f;

#pragma unroll
    for (int mi = 0; mi < 2; ++mi) {
        v8f t0 = mi ? acc10 : acc00;
        v8f t1 = mi ? acc11 : acc01;
        int rbase = row0 + mi * 16 + mbase;
#pragma unroll
        for (int r = 0; r < 8; ++r) {
            size_t off0 = (size_t)(rbase + r) * kout + col_g0;
            size_t off1 = (size_t)(rbase + r) * kout + col_g1;
            float v0 = t0[r] + bv0;
            float v1 = t1[r] + bv1;
            if (accum) { v0 += C[off0]; v1 += C[off1]; }
            if (relu)  { v0 = fmaxf(v0, 0.0f); v1 = fmaxf(v1, 0.0f); }
            C[off0] = v0;
            C[off1] = v1;
        }
    }
}

// ---------------------------------------------------------------------------
// GAT kernels
// ---------------------------------------------------------------------------
// a_s[i,h] = sum_c hg[i,h,c]*att_src[h,c] ; a_d likewise. One wave per (node,head).
__global__ void gat_scores_kernel(const float* __restrict__ hg,
                                  const float* __restrict__ att_src,
                                  const float* __restrict__ att_dst,
                                  float* __restrict__ a_s, float* __restrict__ a_d, int N) {
    int wave = blockIdx.x * (blockDim.x / WAVE) + (threadIdx.x / WAVE);
    int lane = threadIdx.x & (WAVE - 1);
    if (wave >= N * 8) return;
    int node = wave >> 3, head = wave & 7;
    const float* v  = hg + (size_t)node * 1024 + head * 128;
    const float* ws = att_src + head * 128;
    const float* wd = att_dst + head * 128;
    float ss = 0.f, sd = 0.f;
    for (int c = lane; c < 128; c += WAVE) {
        float x = v[c];
        ss += x * ws[c];
        sd += x * wd[c];
    }
    for (int o = WAVE / 2; o >= 1; o >>= 1) {
        ss += __shfl_xor(ss, o, WAVE);
        sd += __shfl_xor(sd, o, WAVE);
    }
    if (lane == 0) { a_s[wave] = ss; a_d[wave] = sd; }
}

__device__ inline float atomic_max_f32(float* addr, float val) {
    int* ai = (int*)addr;
    int old = __float_as_int(*addr);
    while (__int_as_float(old) < val) {
        int assumed = old;
        old = atomicCAS(ai, assumed, __float_as_int(val));
        if (old == assumed) break;
    }
    return __int_as_float(old);
}

__device__ inline void edge_sd(const long long* src, const long long* dst, int E,
                               int e, int& s, int& d) {
    if (e < E) { s = (int)src[e]; d = (int)dst[e]; }
    else       { s = e - E;       d = e - E; }        // self-loops appended
}

// segment max of leaky_relu(a_s[src]+a_d[dst]) per (dst, head)
__global__ void gat_edge_max_kernel(const long long* __restrict__ src,
                                    const long long* __restrict__ dst, int E, int N,
                                    const float* __restrict__ a_s,
                                    const float* __restrict__ a_d,
                                    float* __restrict__ mmax) {
    int Ef = E + N;
    size_t total = (size_t)Ef * 8;
    size_t stride = (size_t)gridDim.x * blockDim.x;
    for (size_t i = (size_t)blockIdx.x * blockDim.x + threadIdx.x; i < total; i += stride) {
        int e = (int)(i >> 3), h = (int)(i & 7);
        int s, d; edge_sd(src, dst, E, e, s, d);
        float x = a_s[s * 8 + h] + a_d[d * 8 + h];
        x = (x > 0.f) ? x : 0.2f * x;                 // leaky_relu(0.2)
        atomic_max_f32(&mmax[d * 8 + h], x);
    }
}

// ew = exp(e - m[dst]) ; denom[dst,h] += ew
__global__ void gat_edge_exp_kernel(const long long* __restrict__ src,
                                    const long long* __restrict__ dst, int E, int N,
                                    const float* __restrict__ a_s,
                                    const float* __restrict__ a_d,
                                    const float* __restrict__ mmax,
                                    float* __restrict__ ew, float* __restrict__ denom) {
    int Ef = E + N;
    size_t total = (size_t)Ef * 8;
    size_t stride = (size_t)gridDim.x * blockDim.x;
    for (size_t i = (size_t)blockIdx.x * blockDim.x + threadIdx.x; i < total; i += stride) {
        int e = (int)(i >> 3), h = (int)(i & 7);
        int s, d; edge_sd(src, dst, E, e, s, d);
        float x = a_s[s * 8 + h] + a_d[d * 8 + h];
        x = (x > 0.f) ? x : 0.2f * x;
        float w = expf(x - mmax[d * 8 + h]);
        ew[i] = w;
        atomicAdd(&denom[d * 8 + h], w);
    }
}

// gat[dst, h, :] += alpha * hg[src, h, :]. One wave per (edge, head), 4 chans/lane.
__global__ void gat_scatter_kernel(const long long* __restrict__ src,
                                   const long long* __restrict__ dst, int E, int N,
                                   const float* __restrict__ ew,
                                   const float* __restrict__ denom,
                                   const float* __restrict__ hg,
                                   float* __restrict__ gat) {
    int wave = blockIdx.x * (blockDim.x / WAVE) + (threadIdx.x / WAVE);
    int lane = threadIdx.x & (WAVE - 1);
    int Ef = E + N;
    if (wave >= Ef * 8) return;
    int e = wave >> 3, h = wave & 7;
    int s, d; edge_sd(src, dst, E, e, s, d);
    float alpha = ew[(size_t)e * 8 + h] / denom[d * 8 + h];
    const float* vs = hg  + (size_t)s * 1024 + h * 128;
    float*       vd = gat + (size_t)d * 1024 + h * 128;
#pragma unroll
    for (int j = 0; j < 4; ++j) {
        int c = lane + j * WAVE;
        atomicAdd(&vd[c], alpha * vs[c]);
    }
}

// ---------------------------------------------------------------------------
// GCN scatter: out[dst,:] += dinv[src]*dinv[dst] * hw[src,:]. One wave per edge.
// ---------------------------------------------------------------------------
__global__ void gcn_scatter_kernel(const long long* __restrict__ src,
                                   const long long* __restrict__ dst, int E, int N,
                                   const float* __restrict__ dinv,
                                   const float* __restrict__ hw,
                                   float* __restrict__ out, int F) {
    int wave = blockIdx.x * (blockDim.x / WAVE) + (threadIdx.x / WAVE);
    int lane = threadIdx.x & (WAVE - 1);
    int Ef = E + N;
    if (wave >= Ef) return;
    int s, d; edge_sd(src, dst, E, wave, s, d);
    float nrm = dinv[s] * dinv[d];
    for (int c = lane; c < F; c += WAVE)
        atomicAdd(&out[(size_t)d * F + c], nrm * hw[(size_t)s * F + c]);
}

// ---------------------------------------------------------------------------
// log_softmax over 32 classes. One wave per node, one lane per class.
// ---------------------------------------------------------------------------
__global__ void log_softmax_kernel(const float* __restrict__ in, float* __restrict__ out, int N) {
    int wave = blockIdx.x * (blockDim.x / WAVE) + (threadIdx.x / WAVE);
    int lane = threadIdx.x & (WAVE - 1);
    if (wave >= N) return;
    float x = in[(size_t)wave * 32 + lane];
    float m = x;
    for (int o = WAVE / 2; o >= 1; o >>= 1) m = fmaxf(m, __shfl_xor(m, o, WAVE));
    float ex = expf(x - m), sum = ex;
    for (int o = WAVE / 2; o >= 1; o >>= 1) sum += __shfl_xor(sum, o, WAVE);
    out[(size_t)wave * 32 + lane] = x - m - logf(sum);
}

// ---------------------------------------------------------------------------
// Host side
// ---------------------------------------------------------------------------
static inline int grid_for(size_t work, int block) {
    size_t g = (work + block - 1) / block;
    if (g > 1048576) g = 1048576;   // grid-stride covers the rest
    return (int)g;
}

static inline void launch_gemm(const float* A, const float* W, const float* bias, float* C,
                               int nrows, int kin, int kout, int accum, int relu,
                               hipStream_t stream) {
    int waves = (nrows / 32) * (kout / 32);
    int blocks = (waves * WAVE + 255) / 256;
    gemm_f32_wmma_kernel<<<blocks, 256, 0, stream>>>(A, W, bias, C, nrows, kin, kout, accum, relu);
}

extern "C" void kernel_launch(void* const* d_in, const int* in_sizes, int n_in,
                              void* d_out, int out_size, void* d_ws, size_t ws_size,
                              hipStream_t stream) {
    const float*     x       = (const float*)d_in[0];
    const long long* ei      = (const long long*)d_in[1];   // int64 per reference
    const float*     W_l1    = (const float*)d_in[2];
    const float*     b_l1    = (const float*)d_in[3];
    const float*     W_r1    = (const float*)d_in[4];
    const float*     W_l2    = (const float*)d_in[5];
    const float*     b_l2    = (const float*)d_in[6];
    const float*     W_r2    = (const float*)d_in[7];
    const float*     W_g     = (const float*)d_in[8];
    const float*     att_src = (const float*)d_in[9];
    const float*     att_dst = (const float*)d_in[10];
    const float*     b_g     = (const float*)d_in[11];
    const float*     W_c1    = (const float*)d_in[12];
    const float*     b_c1    = (const float*)d_in[13];
    const float*     W_c2    = (const float*)d_in[14];
    const float*     b_c2    = (const float*)d_in[15];
    const float*     W_res   = (const float*)d_in[16];
    const float*     b_res   = (const float*)d_in[17];
    float*           out     = (float*)d_out;

    const int N  = in_sizes[0] / 256;      // 20000
    const int E  = in_sizes[1] / 2;        // 160000
    const int Ef = E + N;                  // with self-loops
    const long long* src = ei;
    const long long* dst = ei + E;

    // ---- workspace layout (floats) ----
    float* ws = (float*)d_ws;
    float* cnt   = ws;                         ws += N;             // in-degree, no loops
    float* dinv  = ws;                         ws += N;
    float* agg   = ws;                         ws += (size_t)N * 256;  // SAGE agg/mean (reused)
    float* res_x = ws;                         ws += (size_t)N * 64;
    float* h1    = ws;                         ws += (size_t)N * 64;
    float* h2    = ws;                         ws += (size_t)N * 128;
    float* hg    = ws;                         ws += (size_t)N * 1024;
    float* gat   = ws;                         ws += (size_t)N * 1024;  // also h3 after relu
    float* a_s   = ws;                         ws += (size_t)N * 8;
    float* a_d   = ws;                         ws += (size_t)N * 8;
    float* mmax  = ws;                         ws += (size_t)N * 8;
    float* denom = ws;                         ws += (size_t)N * 8;
    float* ew    = ws;                         ws += (size_t)Ef * 8;
    float* hw1   = ws;                         ws += (size_t)N * 64;
    float* out1  = ws;                         ws += (size_t)N * 64;    // h4
    float* hw2   = ws;                         ws += (size_t)N * 32;
    float* out2  = ws;                         ws += (size_t)N * 32;

    const int B = 256;

    // degrees (original edges)
    fill_val_kernel<<<grid_for(N, B), B, 0, stream>>>(cnt, N, 0.0f);
    degree_kernel<<<grid_for(E, B), B, 0, stream>>>(dst, E, cnt);

    // residual: res_x = x @ W_res + b_res
    launch_gemm(x, W_res, b_res, res_x, N, 256, 64, 0, 0, stream);

    // ---- SAGE1 ----
    fill_val_kernel<<<grid_for((size_t)N * 256, B), B, 0, stream>>>(agg, (size_t)N * 256, 0.0f);
    edge_scatter_sum_kernel<<<grid_for((size_t)E * 256, B), B, 0, stream>>>(src, dst, E, x, agg, 256);
    mean_div_kernel<<<grid_for((size_t)N * 256, B), B, 0, stream>>>(agg, cnt, (size_t)N * 256, 256);
    launch_gemm(agg, W_l1, b_l1, h1, N, 256, 64, 0, 0, stream);      // mean @ W_l1 + b
    launch_gemm(x,   W_r1, nullptr, h1, N, 256, 64, 1, 1, stream);   // += x @ W_r1, relu
    add_kernel<<<grid_for((size_t)N * 64, B), B, 0, stream>>>(h1, res_x, (size_t)N * 64); // + res_x

    // ---- SAGE2 ----
    fill_val_kernel<<<grid_for((size_t)N * 64, B), B, 0, stream>>>(agg, (size_t)N * 64, 0.0f);
    edge_scatter_sum_kernel<<<grid_for((size_t)E * 64, B), B, 0, stream>>>(src, dst, E, h1, agg, 64);
    mean_div_kernel<<<grid_for((size_t)N * 64, B), B, 0, stream>>>(agg, cnt, (size_t)N * 64, 64);
    launch_gemm(agg, W_l2, b_l2, h2, N, 64, 128, 0, 0, stream);
    launch_gemm(h1,  W_r2, nullptr, h2, N, 64, 128, 1, 1, stream);   // relu fused

    // ---- GAT ----
    launch_gemm(h2, W_g, nullptr, hg, N, 128, 1024, 0, 0, stream);   // hg = h2 @ W_g
    gat_scores_kernel<<<(N * 8 + 7) / 8, 256, 0, stream>>>(hg, att_src, att_dst, a_s, a_d, N);
    fill_val_kernel<<<grid_for((size_t)N * 8, B), B, 0, stream>>>(mmax, (size_t)N * 8, -3.4e38f);
    fill_val_kernel<<<grid_for((size_t)N * 8, B), B, 0, stream>>>(denom, (size_t)N * 8, 0.0f);
    fill_val_kernel<<<grid_for((size_t)N * 1024, B), B, 0, stream>>>(gat, (size_t)N * 1024, 0.0f);
    gat_edge_max_kernel<<<grid_for((size_t)Ef * 8, B), B, 0, stream>>>(src, dst, E, N, a_s, a_d, mmax);
    gat_edge_exp_kernel<<<grid_for((size_t)Ef * 8, B), B, 0, stream>>>(src, dst, E, N, a_s, a_d, mmax, ew, denom);
    gat_scatter_kernel<<<(Ef * 8 + 7) / 8, 256, 0, stream>>>(src, dst, E, N, ew, denom, hg, gat);
    bias_relu_kernel<<<grid_for((size_t)N * 1024, B), B, 0, stream>>>(gat, b_g, (size_t)N * 1024, 1024);

    // ---- GCN ----
    dinv_kernel<<<grid_for(N, B), B, 0, stream>>>(dinv, cnt, N);
    launch_gemm(gat, W_c1, nullptr, hw1, N, 1024, 64, 0, 0, stream);
    init_bias_kernel<<<grid_for((size_t)N * 64, B), B, 0, stream>>>(out1, b_c1, (size_t)N * 64, 64);
    gcn_scatter_kernel<<<(Ef + 7) / 8, 256, 0, stream>>>(src, dst, E, N, dinv, hw1, out1, 64);

    launch_gemm(out1, W_c2, nullptr, hw2, N, 64, 32, 0, 0, stream);
    init_bias_kernel<<<grid_for((size_t)N * 32, B), B, 0, stream>>>(out2, b_c2, (size_t)N * 32, 32);
    gcn_scatter_kernel<<<(Ef + 7) / 8, 256, 0, stream>>>(src, dst, E, N, dinv, hw2, out2, 32);

    // ---- log_softmax -> d_out ----
    log_softmax_kernel<<<(N + 7) / 8, 256, 0, stream>>>(out2, out, N);
}